// Ive_53712861004277
// MI455X (gfx1250) — compile-verified
//
#include <hip/hip_runtime.h>

// ============================================================================
// ive(v=49.5, z), fp32, 33.5M elements — CDNA5 WMMA-based series evaluation.
//
// S = sum_k x^k / prod_{i<=k} d(i),  x = z^2/4, d(i) = i*(v+i).
// One 64-term scaled Estrin group is fp32-exact over z in [0.5, 99.5]:
// worst-case tail (z=99.5): term_64 ~ e^16.8, S ~ e^40, geometric factor 1.5
//   =>  tail/S ~ 1e-10 << fp32 eps (1.2e-7), monotone in z.
//
// The 16x4 inner products of the Estrin block run on the matrix pipe:
//   A (16x4)  = constant scaled coefficients A[j] = prod(sigma/d(i)), j=4m+k
//   B (4x16)  = per-element powers [1, w, w^2, w^3],  w = x/sigma  (<= 0.95)
//   D (16x16) = D[m][elem] = sum_k A[4m+k] w^k   -> V_WMMA_F32_16X16X4_F32
// then S = sum_{m=0}^{15} y^m D[m],  y = w^4, via an 8-deep Horner per lane
// pair (D rows 0-7 live in lane i, rows 8-15 in lane i+16) + one shfl_xor(16).
//
// Prefactor in log2 domain on raw HW transcendentals:
//   r = exp2( 49.5*log2(z) - z*log2e - (lgamma(50.5)+49.5*ln2... folded) ) * S
// Any fp32 log's ~1-ulp error is amplified 49.5x (~2e-5 rel) — the fp32
// reference has the same floor — so v_log_f32/v_exp_f32 lose nothing while
// costing 2 TRANS + 2 FMA + 1 MUL total (TRANS co-executes with the Horner).
//
// 64 elements per wave-iteration (2 independent sets per lane, b64 I/O);
// wave-uniform loop + uniform full/tail branch keeps EXEC all-ones around
// the WMMAs as the ISA requires.
// ============================================================================

typedef float v2f __attribute__((ext_vector_type(2)));
typedef float v8f __attribute__((ext_vector_type(8)));

constexpr double V_ORD   = 49.5;
constexpr double LGAMMA  = 146.51925549072064;   // lgamma(50.5)
constexpr double LOG2E   = 1.4426950408889634;
// lp2 = 49.5*log2(z) - LOG2E*z - (LGAMMA*LOG2E + 49.5)   [the +49.5 folds the
// log2(z/2) = log2(z) - 1 shift];  result = exp2(lp2) * S
constexpr float  C_NEG   = (float)(-(LGAMMA * LOG2E + 49.5));
constexpr float  NLOG2E  = (float)(-LOG2E);

constexpr double dden(int i) { return (double)i * (V_ORD + (double)i); }

struct Tables {
  float a0[32];      // WMMA A operand, VGPR0 per lane: lanes<16 K0, lanes>=16 K2
  float a1[32];      // WMMA A operand, VGPR1 per lane: lanes<16 K1, lanes>=16 K3
  float inv_sigma;   // 1/sigma
};

constexpr Tables make_tables() {
  Tables t{};
  const double sigma = dden(32);          // mid-group scale keeps A in fp32 range
  double A[64] = {};
  A[0] = 1.0;
  double p = 1.0;
  for (int jj = 1; jj < 64; ++jj) { p *= sigma / dden(jj); A[jj] = p; }
  t.inv_sigma = (float)(1.0 / sigma);
  // ISA f32 A(16x4) layout: lanes 0-15: VGPR0=K0, VGPR1=K1;
  //                         lanes 16-31: VGPR0=K2, VGPR1=K3;  row M = lane&15
  for (int lane = 0; lane < 32; ++lane) {
    const int m = lane & 15;
    const int kb = (lane < 16) ? 0 : 2;
    t.a0[lane] = (float)A[4 * m + kb + 0];
    t.a1[lane] = (float)A[4 * m + kb + 1];
  }
  return t;
}

__device__ __constant__ Tables g_tab = make_tables();

__global__ __launch_bounds__(256) void ive_wmma_kernel(
    const float* __restrict__ zin, float* __restrict__ out, int n) {
  const int lane = threadIdx.x & 31;
  const bool lo  = (lane < 16);

  // Per-lane constant A operand, hoisted out of the loop.
  v2f a;
  a.x = g_tab.a0[lane];
  a.y = g_tab.a1[lane];
  const float isg = g_tab.inv_sigma;

  // 64-term scaled-series sum for one element set (one element per lane).
  auto series64 = [&](float z) -> float {
    const float x   = 0.25f * z * z;
    const float w   = x * isg;
    const float ws  = __shfl_xor(w, 16, 32);   // partner half-wave's w
    const float w2  = w * w,   w3  = w2 * w;
    const float ws2 = ws * ws, ws3 = ws2 * ws;
    const float y   = w2 * w2;                 // w^4
    const float ys  = ws2 * ws2;
    const float y2  = y * y;                   // w^8
    const float y4  = y2 * y2;                 // w^16
    const float y8  = y4 * y4;                 // w^32

    // B (4x16), ISA layout: VGPR0 = {K0 | K2}, VGPR1 = {K1 | K3}.
    // WMMA#1 columns = elements of lanes 0-15; WMMA#2 = lanes 16-31.
    v2f bA, bB;
    bA.x = lo ? 1.0f : ws2;   bA.y = lo ? w  : ws3;
    bB.x = lo ? 1.0f : w2;    bB.y = lo ? ws : w3;

    v8f cz = {};
    v8f d1 = __builtin_amdgcn_wmma_f32_16x16x4_f32(
        false, a, false, bA, (short)0, cz, false, false);
    v8f d2 = __builtin_amdgcn_wmma_f32_16x16x4_f32(
        false, a, false, bB, (short)0, cz, false, false);

    // D layout: VGPR r holds row M=r (lanes 0-15) / M=r+8 (lanes 16-31).
    const float ya = lo ? y : ys;      // y of the element whose partial we hold
    const float yb = lo ? ys : y;
    float h1 = d1[7], h2 = d2[7];
#pragma unroll
    for (int m = 6; m >= 0; --m) {
      h1 = fmaf(h1, ya, d1[m]);
      h2 = fmaf(h2, yb, d2[m]);
    }
    const float h1s = __shfl_xor(h1, 16, 32);
    const float h2s = __shfl_xor(h2, 16, 32);
    const float Slow  = lo ? h1  : h2s;
    const float Shigh = lo ? h1s : h2;
    return fmaf(y8, Shigh, Slow);              // 64-term sum S
  };

  auto prefactor = [&](float z) -> float {
    // exp2(49.5*log2(z) + (-log2e*z + C)); raw v_log_f32 / v_exp_f32.
    const float l2 = __builtin_amdgcn_logf(z);                 // log2(z)
    const float lp = fmaf(49.5f, l2, fmaf(NLOG2E, z, C_NEG));
    return __builtin_amdgcn_exp2f(lp);                         // 0 on underflow,
  };                                                           // same as ref fp32

  // Wave-uniform grid-stride: 64 consecutive elements per wave-iteration,
  // two per lane (b64 I/O). Loop bound and full/tail check are uniform, so
  // EXEC is all-ones around every WMMA.
  const unsigned wave  = blockIdx.x * (blockDim.x >> 5) + (threadIdx.x >> 5);
  const unsigned nwave = gridDim.x * (blockDim.x >> 5);
  for (long long base = (long long)wave * 64; base < n;
       base += (long long)nwave * 64) {
    const int i0 = (int)base + 2 * lane;
    const int i1 = i0 + 1;
    const bool full = (base + 63) < (long long)n;  // uniform across the wave

    float z0, z1;
    if (full) {
      const float2 v = *(const float2*)(zin + i0);
      z0 = v.x; z1 = v.y;
    } else {
      z0 = zin[i0 < n ? i0 : (n - 1)];
      z1 = zin[i1 < n ? i1 : (n - 1)];
    }

    // Two independent element sets -> two independent WMMA/Horner chains.
    const float s0 = series64(z0);
    const float s1 = series64(z1);
    const float r0 = prefactor(z0) * s0;
    const float r1 = prefactor(z1) * s1;

    if (full) {
      float2 v; v.x = r0; v.y = r1;
      *(float2*)(out + i0) = v;
    } else {
      if (i0 < n) out[i0] = r0;
      if (i1 < n) out[i1] = r1;
    }
  }
}

extern "C" void kernel_launch(void* const* d_in, const int* in_sizes, int n_in,
                              void* d_out, int out_size, void* d_ws,
                              size_t ws_size, hipStream_t stream) {
  (void)in_sizes; (void)n_in; (void)d_ws; (void)ws_size;
  const float* z = (const float*)d_in[0];
  float* out = (float*)d_out;
  const int n = out_size;                 // 4096*8192 for the reference shapes

  const int threads = 256;                // 8 wave32s per block
  long long waves_needed = ((long long)n + 63) / 64;
  long long blocks = (waves_needed + 7) / 8;
  if (blocks > 4096) blocks = 4096;       // grid-stride beyond this
  if (blocks < 1) blocks = 1;
  ive_wmma_kernel<<<(int)blocks, threads, 0, stream>>>(z, out, n);
}